// gap_layer_6399501271885
// MI455X (gfx1250) — compile-verified
//
#include <hip/hip_runtime.h>
#include <hip/hip_bf16.h>
#include <math.h>

// CDNA5 / gfx1250. wave32.
typedef __attribute__((ext_vector_type(2))) float v2f;
typedef __attribute__((ext_vector_type(8))) float v8f;

#define KNN 20
#define NPTS 4096

// Monotone map: float -> uint32 such that unsigned compare == float compare.
__device__ __forceinline__ unsigned sortable_f32(float f) {
    unsigned u = __float_as_uint(f);
    return (u & 0x80000000u) ? ~u : (u ^ 0x80000000u);
}

// ---------------------------------------------------------------------------
// Kernel 1: fused Gram-matrix (WMMA f32 16x16x4) + streaming top-20 selection.
// One wave (32 lanes) per 32 query points. Per column tile of 16 candidates:
//   two v_wmma_f32_16x16x4_f32 -> 32x16 inner products,
//   negd = 2*inner - xx_row - xx_col staged via 2KB LDS,
//   each lane owns one query row; top-20 kept as sorted u64 keys in VGPRs,
//   key = (sortable(negdist) << 32) | ~idx  => single u64 compare handles
//   both descending-value ordering and lax.top_k's lower-index tie-break.
// ---------------------------------------------------------------------------
__global__ __launch_bounds__(32) void knn_kernel(const float* __restrict__ x,
                                                 int* __restrict__ idx_out,
                                                 int N) {
    __shared__ float tile[32 * 16];

    const int lane = threadIdx.x;
    const int b = blockIdx.y;
    const int rowBase = blockIdx.x * 32;
    const float* xb = x + (size_t)b * 3 * N;

    const int half = lane >> 4;   // 0: K=0,1  |  1: K=2,3(pad)
    const int l16 = lane & 15;

    // A operands for the two 16-row blocks (rows rowBase..+15, +16..+31)
    v2f aA, aB;
    {
        int m = rowBase + l16;
        if (half == 0) { aA.x = xb[0 * N + m]; aA.y = xb[1 * N + m]; }
        else           { aA.x = xb[2 * N + m]; aA.y = 0.0f; }
        int m2 = m + 16;
        if (half == 0) { aB.x = xb[0 * N + m2]; aB.y = xb[1 * N + m2]; }
        else           { aB.x = xb[2 * N + m2]; aB.y = 0.0f; }
    }

    // ||x_m||^2 for the 8 D-rows this lane produces per WMMA result
    float xxA[8], xxB[8];
    const int mb = rowBase + (half ? 8 : 0);
#pragma unroll
    for (int r = 0; r < 8; ++r) {
        int m = mb + r;
        float px = xb[0 * N + m], py = xb[1 * N + m], pz = xb[2 * N + m];
        xxA[r] = px * px + py * py + pz * pz;
        m += 16;
        px = xb[0 * N + m]; py = xb[1 * N + m]; pz = xb[2 * N + m];
        xxB[r] = px * px + py * py + pz * pz;
    }

    // Register-resident sorted (descending) top-20 keys.
    unsigned long long top[KNN];
#pragma unroll
    for (int j = 0; j < KNN; ++j) top[j] = 0ull;

    for (int cb = 0; cb < N; cb += 16) {
        const int cand = cb + l16;
        const float cx = xb[0 * N + cand];
        const float cy = xb[1 * N + cand];
        const float cz = xb[2 * N + cand];
        const float xxn = cx * cx + cy * cy + cz * cz;

        // prefetch next candidate tile into near caches
        if (cb + 16 < N) __builtin_prefetch(xb + cb + 16 + l16, 0, 3);

        v2f bv;
        if (half == 0) { bv.x = cx; bv.y = cy; }
        else           { bv.x = cz; bv.y = 0.0f; }

        v8f c1 = {};
        v8f c2 = {};
        c1 = __builtin_amdgcn_wmma_f32_16x16x4_f32(false, aA, false, bv,
                                                   (short)0, c1, false, false);
        c2 = __builtin_amdgcn_wmma_f32_16x16x4_f32(false, aB, false, bv,
                                                   (short)0, c2, false, false);

        const int lr = half ? 8 : 0;
#pragma unroll
        for (int r = 0; r < 8; ++r) {
            tile[(lr + r) * 16 + l16]      = 2.0f * c1[r] - xxA[r] - xxn;
            tile[(16 + lr + r) * 16 + l16] = 2.0f * c2[r] - xxB[r] - xxn;
        }
        __syncthreads();

        // selection: lane owns local row `lane`; keep this loop rolled so the
        // hot code stays small (only the 20-step insert is unrolled so top[]
        // remains in VGPRs).
#pragma unroll 1
        for (int j = 0; j < 16; ++j) {
            float v = tile[lane * 16 + j];
            unsigned long long key =
                ((unsigned long long)sortable_f32(v) << 32) |
                (unsigned)(~(cb + j));
            if (key > top[KNN - 1]) {
                unsigned long long k = key;
#pragma unroll
                for (int s = 0; s < KNN; ++s) {
                    bool better = (k > top[s]);
                    unsigned long long t = top[s];
                    top[s] = better ? k : t;    // branchless: v_cndmask pairs
                    k      = better ? t : k;
                }
            }
        }
        __syncthreads();
    }

    const int n = rowBase + lane;
    int* dst = idx_out + ((size_t)b * N + n) * KNN;
#pragma unroll
    for (int j = 0; j < KNN; ++j)
        dst[j] = (int)(~(unsigned)(top[j] & 0xFFFFFFFFull));
}

// ---------------------------------------------------------------------------
// Kernel 2: gather neighbors, edge MLPs (BN folded), attention softmax over K
// within a wave (shfl_xor, wave32), ELU, and both outputs.
// One wave per point; lane = neighbor index k (0..19 active).
// ---------------------------------------------------------------------------
__global__ __launch_bounds__(256) void gap_kernel(
    const float* __restrict__ x, const int* __restrict__ knn,
    const float* __restrict__ w1,
    const float* __restrict__ g1, const float* __restrict__ be1,
    const float* __restrict__ m1, const float* __restrict__ v1,
    const float* __restrict__ w2, const float* __restrict__ b2,
    const float* __restrict__ g2, const float* __restrict__ be2,
    const float* __restrict__ m2, const float* __restrict__ v2,
    const float* __restrict__ w3, const float* __restrict__ b3,
    const float* __restrict__ g3, const float* __restrict__ be3,
    const float* __restrict__ m3, const float* __restrict__ v3,
    float* __restrict__ ret, float* __restrict__ edgef, int N) {
    const int wave = threadIdx.x >> 5;
    const int lane = threadIdx.x & 31;
    const int p = blockIdx.x * 8 + wave;       // global point id
    const int b = p / N;
    const int n = p - b * N;
    const float* xb = x + (size_t)b * 3 * N;

    const float sx = xb[0 * N + n], sy = xb[1 * N + n], sz = xb[2 * N + n];
    float dx = 0.0f, dy = 0.0f, dz = 0.0f;
    const bool act = (lane < KNN);
    if (act) {
        int nb = knn[(size_t)p * KNN + lane];
        dx = sx - xb[0 * N + nb];
        dy = sy - xb[1 * N + nb];
        dz = sz - xb[2 * N + nb];
    }

    const float sc3 = g3[0] * rsqrtf(v3[0] + 1e-5f);
    const float sh3 = be3[0] - m3[0] * sc3;
    const float bias3 = b3[0];

    float e[16];
    float sself = 0.0f, sneib = 0.0f;
#pragma unroll
    for (int c = 0; c < 16; ++c) {
        const float sc1 = g1[c] * rsqrtf(v1[c] + 1e-5f);
        float f1 = dx * w1[c * 3] + dy * w1[c * 3 + 1] + dz * w1[c * 3 + 2];
        f1 = fmaxf(f1 * sc1 + (be1[c] - m1[c] * sc1), 0.0f);
        sself += w3[c] * f1;

        const float sc2 = g2[c] * rsqrtf(v2[c] + 1e-5f);
        float ec = dx * w2[c * 3] + dy * w2[c * 3 + 1] + dz * w2[c * 3 + 2] + b2[c];
        ec = fmaxf(ec * sc2 + (be2[c] - m2[c] * sc2), 0.0f);
        e[c] = ec;
        sneib += w3[c] * ec;
    }

    const float satt = fmaxf((sself + bias3) * sc3 + sh3, 0.0f);
    const float natt = fmaxf((sneib + bias3) * sc3 + sh3, 0.0f);
    float logit = satt + natt;
    logit = (logit > 0.0f) ? logit : 0.01f * logit;   // leaky_relu(0.01)
    if (!act) logit = -3.0e38f;

    // softmax over K within the wave (wave32 shfl_xor reductions)
    float mx = logit;
#pragma unroll
    for (int off = 16; off >= 1; off >>= 1)
        mx = fmaxf(mx, __shfl_xor(mx, off, 32));
    float ex = act ? expf(logit - mx) : 0.0f;
    float den = ex;
#pragma unroll
    for (int off = 16; off >= 1; off >>= 1)
        den += __shfl_xor(den, off, 32);
    const float coef = ex / den;

#pragma unroll
    for (int c = 0; c < 16; ++c) {
        float v = coef * e[c];
#pragma unroll
        for (int off = 16; off >= 1; off >>= 1)
            v += __shfl_xor(v, off, 32);
        if (lane == 0) {
            float o = (v > 0.0f) ? v : (expf(v) - 1.0f);   // ELU
            ret[(size_t)p * 16 + c] = o;
        }
        if (act)
            edgef[(((size_t)b * 16 + c) * N + n) * KNN + lane] = e[c];
    }
}

// ---------------------------------------------------------------------------
extern "C" void kernel_launch(void* const* d_in, const int* in_sizes, int n_in,
                              void* d_out, int out_size, void* d_ws, size_t ws_size,
                              hipStream_t stream) {
    const float* x   = (const float*)d_in[0];
    const float* w1  = (const float*)d_in[1];
    const float* g1  = (const float*)d_in[2];
    const float* be1 = (const float*)d_in[3];
    const float* m1  = (const float*)d_in[4];
    const float* v1  = (const float*)d_in[5];
    const float* w2  = (const float*)d_in[6];
    const float* b2  = (const float*)d_in[7];
    const float* g2  = (const float*)d_in[8];
    const float* be2 = (const float*)d_in[9];
    const float* m2  = (const float*)d_in[10];
    const float* v2  = (const float*)d_in[11];
    const float* w3  = (const float*)d_in[12];
    const float* b3  = (const float*)d_in[13];
    const float* g3  = (const float*)d_in[14];
    const float* be3 = (const float*)d_in[15];
    const float* m3  = (const float*)d_in[16];
    const float* v3  = (const float*)d_in[17];

    const int N = NPTS;
    const int B = in_sizes[0] / (3 * N);   // = 8

    int* knn = (int*)d_ws;                 // (B, N, 20) ints = 2.6 MB scratch

    dim3 grid1(N / 32, B);
    knn_kernel<<<grid1, 32, 0, stream>>>(x, knn, N);

    float* ret   = (float*)d_out;                       // (B, N, 1, 16)
    float* edgef = ret + (size_t)B * N * 16;            // (B, 16, N, 20)
    gap_kernel<<<dim3((B * N) / 8), 256, 0, stream>>>(
        x, knn, w1, g1, be1, m1, v1, w2, b2, g2, be2, m2, v2,
        w3, b3, g3, be3, m3, v3, ret, edgef, N);
}